// EPGSimulationSliceProfile_12781822672904
// MI455X (gfx1250) — compile-verified
//
#include <hip/hip_runtime.h>
#include <hip/hip_bf16.h>
#include <cstdint>

// ---------------------------------------------------------------------------
// EPG slice-profile scan for MI455X (gfx1250, wave32).
//   - 1 wave == 1 sub-slice; lane L holds states k=2L and k=2L+1 (n_states=64).
//   - Rolls (+1 for Fp/Z, -1 for Fm) are in-wave shuffles only.
//   - Pulse tables staged to LDS via gfx1250 async-to-LDS (ASYNCcnt path).
//   - ALL trig hoisted out of the hot loop: lane L precomputes sincos of the
//     flip angles for pulses 4L..4L+3; the loop broadcasts them with 2 shuffles.
//   - History streamed with non-temporal b128 stores (~335 MB -> ~14 us floor).
// ---------------------------------------------------------------------------

typedef float v4f __attribute__((ext_vector_type(4)));
typedef float v2f __attribute__((ext_vector_type(2)));

typedef __attribute__((address_space(1))) int* gptr_i32;
typedef __attribute__((address_space(3))) int* lptr_i32;

#ifndef EPG_ASYNC_LDS
#  if defined(__has_builtin)
#    if __has_builtin(__builtin_amdgcn_global_load_async_to_lds_b32)
#      define EPG_ASYNC_LDS 1
#    else
#      define EPG_ASYNC_LDS 0
#    endif
#  else
#    define EPG_ASYNC_LDS 0
#  endif
#endif

#define MAX_P 128
#define BLOCK_THREADS 256

__global__ __launch_bounds__(BLOCK_THREADS) void epg_scan_kernel(
    const float* __restrict__ flip,    // [P]
    const float* __restrict__ phase,   // [P]
    const float* __restrict__ sliceF,  // [S]
    const float* __restrict__ T1p, const float* __restrict__ T2p,
    const float* __restrict__ TRp, const float* __restrict__ B0p,
    const float* __restrict__ B1p,
    float* __restrict__ out, int P, int S)
{
    // Loop coefficients: c4 = {cos b, sin b, cos 2b, sin 2b}; hc = flip*B1.
    __shared__ v4f   s_c4[MAX_P];
    __shared__ float s_hc[MAX_P];
    __shared__ float s_stage[2 * MAX_P];  // raw flip/phase staging

    const int tid = threadIdx.x;

    // Uniform scalars (uniform addresses -> scalar loads).
    const float T1 = *T1p, T2 = *T2p, TR = *TRp, B0 = *B0p, B1 = *B1p;

    // ---- Stage flip/phase tables into LDS (gfx1250 async-to-LDS path).
    if (tid < P) {
#if EPG_ASYNC_LDS
        __builtin_amdgcn_global_load_async_to_lds_b32(
            (gptr_i32)(flip + tid), (lptr_i32)(&s_stage[tid]), 0, 0);
        __builtin_amdgcn_global_load_async_to_lds_b32(
            (gptr_i32)(phase + tid), (lptr_i32)(&s_stage[MAX_P + tid]), 0, 0);
#else
        s_stage[tid]         = flip[tid];
        s_stage[MAX_P + tid] = phase[tid];
#endif
    }
#if EPG_ASYNC_LDS
#  if defined(__has_builtin) && __has_builtin(__builtin_amdgcn_s_wait_asynccnt)
    __builtin_amdgcn_s_wait_asynccnt(0);
#  else
    asm volatile("s_wait_asynccnt 0" ::: "memory");
#  endif
#endif
    __syncthreads();

    // ---- Build per-pulse phase coefficients (parallel over threads).
    if (tid < P) {
        const float f  = s_stage[tid];
        const float ph = s_stage[MAX_P + tid];
        float sb, cb;
        sincosf(ph, &sb, &cb);
        v4f c4;
        c4.x = cb;
        c4.y = sb;
        c4.z = cb * cb - sb * sb;   // cos(2b)
        c4.w = 2.0f * cb * sb;      // sin(2b)
        s_c4[tid] = c4;
        s_hc[tid] = f * B1;         // full flip coefficient (x sliceF later)
    }
    __syncthreads();   // last barrier; everything below is per-wave private

    const int lane  = tid & 31;
    const int wave  = tid >> 5;
    const int slice = blockIdx.x * (BLOCK_THREADS >> 5) + wave;
    if (slice >= S) return;

    const float sf  = sliceF[slice];
    const float E1  = expf(-TR / T1);
    const float E2  = expf(-TR / T2);
    const float recLane = (lane == 0) ? (1.0f - E1) : 0.0f;  // recovery @ k==0
    const float phi = 2.0f * 3.14159265358979323846f * B0 * TR * 0.001f;
    float pfi, pfr;
    sincosf(phi, &pfi, &pfr);
    // Fold E2 into the B0 phasor: Fp *= E2*pf ; Fm *= E2*conj(pf).
    const float pE2r = E2 * pfr;
    const float pE2i = E2 * pfi;

    // ---- Precompute sincos(flip angle) for ALL pulses, 4 per lane:
    //      lane L owns pulses q = 4L + j (j = 0..3).
    float cA[4], sA[4];
#pragma unroll
    for (int j = 0; j < 4; ++j) {
        const int q = 4 * lane + j;
        const float A = (q < P) ? sf * s_hc[q] : 0.0f;
        sincosf(A, &sA[j], &cA[j]);
    }

    // Carry state: lane holds states k0=2*lane, k1=2*lane+1.
    float Fp0r = 0.f, Fp0i = 0.f, Fp1r = 0.f, Fp1i = 0.f;
    float Fm0r = 0.f, Fm0i = 0.f, Fm1r = 0.f, Fm1i = 0.f;
    float Z0 = (lane == 0) ? 1.0f : 0.0f, Z1 = 0.f;

    const size_t fmBase = (size_t)P * (size_t)S * 128u;  // floats in Fp block
    float* __restrict__ outFp = out;
    float* __restrict__ outFm = out + fmBase;
    float* __restrict__ outZ  = out + 2 * fmBase;

    // One EPG step for pulse p; cAp/sAp are cos/sin of the full flip angle.
    auto epg_iter = [&](int p, float cAp, float sAp) {
        const v4f cc = s_c4[p];
        const float cb = cc.x, sb = cc.y, e2r = cc.z, e2i = cc.w;

        // Half-angle identities from the full-angle sincos.
        const float c2 = 0.5f + 0.5f * cAp;  // cos^2(a/2)
        const float s2 = 0.5f - 0.5f * cAp;  // sin^2(a/2)
        const float cs = 0.5f * sAp;         // sin(a/2)cos(a/2)
        const float ca = cAp;                // c2 - s2

        // ---- RF rotation per state ----
#define EPG_STEP(Fpr, Fpi, Fmr, Fmi, Zv, oFpr, oFpi, oFmr, oFmi, oZ)           \
        {                                                                      \
            const float csZ = cs * (Zv);                                       \
            oFpr = c2 * (Fpr) + s2 * ((Fmr) * e2r + (Fmi) * e2i) - csZ * sb;   \
            oFpi = c2 * (Fpi) + s2 * ((Fmr) * e2i - (Fmi) * e2r) + csZ * cb;   \
            oFmr = s2 * ((Fpr) * e2r - (Fpi) * e2i) + c2 * (Fmr) - csZ * sb;   \
            oFmi = c2 * (Fmi) - s2 * ((Fpr) * e2i + (Fpi) * e2r) - csZ * cb;   \
            oZ   = cs * (((Fpi) - (Fmi)) * cb - ((Fpr) + (Fmr)) * sb)          \
                 + ca * (Zv);                                                  \
        }
        float aFpr, aFpi, aFmr, aFmi, aZ;
        float bFpr, bFpi, bFmr, bFmi, bZ;
        EPG_STEP(Fp0r, Fp0i, Fm0r, Fm0i, Z0, aFpr, aFpi, aFmr, aFmi, aZ)
        EPG_STEP(Fp1r, Fp1i, Fm1r, Fm1i, Z1, bFpr, bFpi, bFmr, bFmi, bZ)
#undef EPG_STEP

        // ---- Relaxation (+ recovery folded into a per-lane constant) ----
        aZ = aZ * E1 + recLane;
        bZ = bZ * E1;

        // ---- E2 * B0 phasor: Fp *= (pE2r,pE2i); Fm *= (pE2r,-pE2i) ----
        const float tAFpr = aFpr * pE2r - aFpi * pE2i;
        const float tAFpi = aFpr * pE2i + aFpi * pE2r;
        const float tAFmr = aFmr * pE2r + aFmi * pE2i;
        const float tAFmi = aFmi * pE2r - aFmr * pE2i;
        const float tBFpr = bFpr * pE2r - bFpi * pE2i;
        const float tBFpi = bFpr * pE2i + bFpi * pE2r;
        const float tBFmr = bFmr * pE2r + bFmi * pE2i;
        const float tBFmi = bFmi * pE2r - bFmr * pE2i;

        // ---- Shifts: Fp/Z roll +1 (k <- k-1), Fm roll -1 (k <- k+1) ----
        const float upFpr = __shfl_up(tBFpr, 1);
        const float upFpi = __shfl_up(tBFpi, 1);
        const float upZ   = __shfl_up(bZ, 1);
        const float dnFmr = __shfl_down(tAFmr, 1);
        const float dnFmi = __shfl_down(tAFmi, 1);

        Fp1r = tAFpr;                          Fp1i = tAFpi;
        Fp0r = (lane == 0) ? 0.0f : upFpr;     Fp0i = (lane == 0) ? 0.0f : upFpi;
        Z1   = aZ;                             Z0   = (lane == 0) ? 0.0f : upZ;
        Fm0r = tBFmr;                          Fm0i = tBFmi;
        Fm1r = (lane == 31) ? 0.0f : dnFmr;    Fm1i = (lane == 31) ? 0.0f : dnFmi;

        // ---- Stream post-shift state to d_out (non-temporal, coalesced) ----
        const size_t row  = (size_t)p * (size_t)S + (size_t)slice;
        const size_t offF = row * 128 + 4 * (size_t)lane;  // complex interleaved
        const size_t offZ = row * 64  + 2 * (size_t)lane;

        v4f vFp = {Fp0r, Fp0i, Fp1r, Fp1i};
        v4f vFm = {Fm0r, Fm0i, Fm1r, Fm1i};
        v2f vZ  = {Z0, Z1};
        __builtin_nontemporal_store(vFp, (v4f*)(outFp + offF));
        __builtin_nontemporal_store(vFm, (v4f*)(outFm + offF));
        __builtin_nontemporal_store(vZ,  (v2f*)(outZ  + offZ));
    };

    // Unroll by 4 so the broadcast register index (p&3) is compile-time;
    // source lane for the sincos broadcast is p>>2.
    for (int p4 = 0; p4 < P; p4 += 4) {
        const int srcLane = p4 >> 2;
#pragma unroll
        for (int j = 0; j < 4; ++j) {
            if (p4 + j < P) {
                const float cAp = __shfl(cA[j], srcLane);
                const float sAp = __shfl(sA[j], srcLane);
                epg_iter(p4 + j, cAp, sAp);
            }
        }
    }
}

extern "C" void kernel_launch(void* const* d_in, const int* in_sizes, int n_in,
                              void* d_out, int out_size, void* d_ws, size_t ws_size,
                              hipStream_t stream) {
    const float* flip   = (const float*)d_in[0];  // nominal_flip_angles_rad_seq [P]
    const float* phase  = (const float*)d_in[1];  // phases_rad_seq [P]
    const float* sliceF = (const float*)d_in[2];  // slice_profile_factors [S]
    const float* T1     = (const float*)d_in[3];
    const float* T2     = (const float*)d_in[4];
    const float* TR     = (const float*)d_in[5];
    /* d_in[6] = TE_ms (unused by the reference math) */
    const float* B0     = (const float*)d_in[7];
    const float* B1     = (const float*)d_in[8];
    /* d_in[9] = n_states; kernel mapping assumes 64 (2 states/lane, wave32) */

    int P = in_sizes[0];
    if (P > MAX_P) P = MAX_P;
    const int S = in_sizes[2];

    const int wavesPerBlock = BLOCK_THREADS / 32;           // 8 sub-slices/block
    const int grid = (S + wavesPerBlock - 1) / wavesPerBlock;

    epg_scan_kernel<<<grid, BLOCK_THREADS, 0, stream>>>(
        flip, phase, sliceF, T1, T2, TR, B0, B1, (float*)d_out, P, S);
}